// _DatasetAwareGaussianFilter_39135742001287
// MI455X (gfx1250) — compile-verified
//
#include <hip/hip_runtime.h>
#include <hip/hip_bf16.h>

typedef float v2f __attribute__((ext_vector_type(2)));
typedef float v4f __attribute__((ext_vector_type(4)));
typedef float v8f __attribute__((ext_vector_type(8)));

#define B_      16
#define C_      8
#define H_      512
#define W_      512
#define RADIUS  24
#define KSIZE   49       // 2*RADIUS+1
#define TILE_Y  16
#define TILE_X  64
#define IN_W    112      // TILE_X + 2*RADIUS
#define IN_W4   28       // IN_W / 4
#define IN_H    64       // TILE_Y + 2*RADIUS
#define STRIDE  116      // padded LDS row stride (floats); STRIDE*4 % 16 == 0

// Fused separable Gaussian: vertical pass (banded GEMM) -> LDS -> horizontal
// pass (banded GEMM, transposed output) using V_WMMA_F32_16X16X4_F32.
__global__ __launch_bounds__(128)
void _gauss_fused_wmma(const float* __restrict__ in,
                       const float* __restrict__ scaling,
                       const int*   __restrict__ dsidx,
                       const float* __restrict__ dssig,
                       float*       __restrict__ out)
{
    __shared__ __align__(16) float s_in[IN_H * STRIDE];   // input halo tile
    __shared__ __align__(16) float s_v [TILE_Y * STRIDE]; // vertically filtered stripe
    __shared__ float s_k [64];              // gaussian taps, zero-padded to 64
    __shared__ float s_norm;

    const int tid  = threadIdx.x;
    const int lane = tid & 31;
    const int wave = tid >> 5;

    const int bc = blockIdx.z;          // b*C_ + c
    const int b  = bc / C_;
    const int y0 = blockIdx.y * TILE_Y;
    const int x0 = blockIdx.x * TILE_X;
    const size_t base = (size_t)bc * (size_t)(H_ * W_);

    // ---- build per-batch gaussian taps in LDS ----
    if (tid < 64) {
        float sg  = dssig[dsidx[b]] * scaling[b];
        float inv = 1.0f / (2.0f * sg * sg);
        float d   = (float)(tid - RADIUS);
        s_k[tid]  = (tid < KSIZE) ? expf(-d * d * inv) : 0.0f;
    }
    __syncthreads();
    if (tid == 0) {
        float s = 0.0f;
        #pragma unroll
        for (int t = 0; t < KSIZE; ++t) s += s_k[t];
        s_norm = 1.0f / s;
    }
    __syncthreads();
    if (tid < 64) s_k[tid] *= s_norm;

    // ---- load input halo into LDS ----
    // Fast path (block-uniform): x-halo fully in range -> aligned float4 loads.
    // (x0 is 64-aligned and RADIUS*4 = 96 bytes is 16B-aligned, so rows are
    //  float4-aligned both in global memory and in LDS.)
    if (x0 >= RADIUS && x0 + (IN_W - RADIUS) <= W_) {
        const float* rowbase = in + base + (size_t)(x0 - RADIUS);
        #pragma unroll
        for (int it = 0; it < (IN_H * IN_W4) / 128; ++it) {  // 14 iters exact
            int i   = tid + it * 128;
            int ry  = i / IN_W4;
            int rx4 = i - ry * IN_W4;
            int gy  = y0 - RADIUS + ry;
            gy = gy < 0 ? 0 : (gy > H_ - 1 ? H_ - 1 : gy);   // row-uniform clamp
            v4f v = *((const v4f*)(rowbase + (size_t)gy * W_) + rx4);
            *(v4f*)(&s_in[ry * STRIDE + rx4 * 4]) = v;
        }
    } else {
        // Edge block columns: per-element clamp on both axes.
        for (int i = tid; i < IN_H * IN_W; i += 128) {
            int ry = i / IN_W;
            int rx = i - ry * IN_W;
            int gy = y0 - RADIUS + ry; gy = gy < 0 ? 0 : (gy > H_ - 1 ? H_ - 1 : gy);
            int gx = x0 - RADIUS + rx; gx = gx < 0 ? 0 : (gx > W_ - 1 ? W_ - 1 : gx);
            s_in[ry * STRIDE + rx] = in[base + (size_t)gy * W_ + gx];
        }
    }
    __syncthreads();   // also covers s_k normalization

    // ---- banded A-matrix chunks (shared by both stages), per ISA A layout:
    //      lanes 0-15 hold (K0,K1), lanes 16-31 hold (K2,K3) of each 16x4 chunk
    const int row  = lane & 15;
    const int koff = (lane >> 4) << 1;
    v2f A[16];
    #pragma unroll
    for (int c = 0; c < 16; ++c) {
        int j0 = 4 * c + koff;
        int d0 = j0 - row;
        int d1 = d0 + 1;
        A[c].x = (d0 >= 0 && d0 < KSIZE) ? s_k[d0] : 0.0f;
        A[c].y = (d1 >= 0 && d1 < KSIZE) ? s_k[d1] : 0.0f;
    }

    // ---- stage 1: vertical filter  V[16 x 112] = M[16x64] x In[64x112] ----
    // 7 column groups of 16, strided across the 4 waves (wave-uniform -> EXEC full).
    for (int h = wave; h < 7; h += 4) {
        v8f acc = {0.f, 0.f, 0.f, 0.f, 0.f, 0.f, 0.f, 0.f};
        const int coln = 16 * h + (lane & 15);
        #pragma unroll
        for (int c = 0; c < 16; ++c) {
            int r0 = 4 * c + koff;      // B chunk: B[kk][n] = s_in[4c+kk][coln]
            v2f bv;
            bv.x = s_in[r0 * STRIDE + coln];
            bv.y = s_in[(r0 + 1) * STRIDE + coln];
            acc = __builtin_amdgcn_wmma_f32_16x16x4_f32(
                      false, A[c], false, bv, (short)0, acc, false, false);
        }
        // D layout: VGPR r = row r (lanes<16) / row 8+r (lanes>=16), col = lane&15
        const int vr0 = (lane >> 4) * 8;
        #pragma unroll
        for (int r = 0; r < 8; ++r)
            s_v[(vr0 + r) * STRIDE + coln] = acc[r];
    }
    __syncthreads();

    // ---- stage 2: horizontal filter, one 16x16 output tile per wave ----
    // O^T[xl][y] = sum_j M[xl][j] * V[y][16g + j]
    {
        const int g  = wave;
        const int yy = lane & 15;
        v8f acc = {0.f, 0.f, 0.f, 0.f, 0.f, 0.f, 0.f, 0.f};
        #pragma unroll
        for (int c = 0; c < 16; ++c) {
            int colb = 16 * g + 4 * c + koff;   // B[kk][y] = s_v[y][16g+4c+kk]
            v2f bv;
            bv.x = s_v[yy * STRIDE + colb];
            bv.y = s_v[yy * STRIDE + colb + 1];
            acc = __builtin_amdgcn_wmma_f32_16x16x4_f32(
                      false, A[c], false, bv, (short)0, acc, false, false);
        }
        // D = O^T: lane owns output row y = lane&15, 8 consecutive columns
        const int oy = y0 + yy;
        const int ox = x0 + 16 * g + (lane >> 4) * 8;
        float* p = out + base + (size_t)oy * W_ + ox;
        v4f lo = {acc[0], acc[1], acc[2], acc[3]};
        v4f hi = {acc[4], acc[5], acc[6], acc[7]};
        *(v4f*)(p)     = lo;
        *(v4f*)(p + 4) = hi;
    }
}

extern "C" void kernel_launch(void* const* d_in, const int* in_sizes, int n_in,
                              void* d_out, int out_size, void* d_ws, size_t ws_size,
                              hipStream_t stream) {
    (void)in_sizes; (void)n_in; (void)out_size; (void)d_ws; (void)ws_size;
    const float* tensor  = (const float*)d_in[0];
    const float* scaling = (const float*)d_in[1];
    const int*   dsidx   = (const int*)  d_in[2];
    const float* dssig   = (const float*)d_in[3];
    float* out = (float*)d_out;

    dim3 grid(W_ / TILE_X, H_ / TILE_Y, B_ * C_);   // 8 x 32 x 128
    dim3 block(128);                                 // 4 waves (wave32)
    _gauss_fused_wmma<<<grid, block, 0, stream>>>(tensor, scaling, dsidx, dssig, out);
}